// gen_proof_23441931501638
// MI455X (gfx1250) — compile-verified
//
#include <hip/hip_runtime.h>
#include <stdint.h>
#include <stddef.h>

// ---------------- field / Montgomery constants (compile-time) ----------------
#define LOG_N 21
#define NN (1u << LOG_N)      // 2097152
#define NTOT (NN * 2u)        // sort length

static constexpr uint64_t PP  = 2013265921ull;   // 15*2^27 + 1
static constexpr uint32_t P32 = 2013265921u;

constexpr uint32_t cmulmod(uint64_t a, uint64_t b) {
  return (uint32_t)((a % PP) * (b % PP) % PP);
}
constexpr uint32_t cpowmod(uint64_t b, uint64_t e) {
  uint64_t r = 1; b %= PP;
  while (e) { if (e & 1) r = r * b % PP; b = b * b % PP; e >>= 1; }
  return (uint32_t)r;
}
constexpr uint32_t mont_np() {           // -P^{-1} mod 2^32 (Newton)
  uint32_t inv = (uint32_t)PP;
  for (int i = 0; i < 5; ++i) inv *= 2u - (uint32_t)PP * inv;
  return (uint32_t)(0u - inv);
}
static constexpr uint32_t NPRIME = mont_np();
static constexpr uint32_t R1 = (uint32_t)((1ull << 32) % PP);   // R mod P
static constexpr uint32_t R2 = cmulmod(R1, R1);                 // R^2 mod P
static constexpr uint32_t ROOT  = cpowmod(31, (PP - 1) / NN);   // 31^960
static constexpr uint32_t IROOT = cpowmod(ROOT, PP - 2);
static constexpr uint32_t NINV  = cpowmod(NN, PP - 2);
static constexpr uint32_t NINV_MONT = cmulmod(NINV, R1);

// ---------------- device field helpers ----------------
__device__ __forceinline__ uint32_t mmul(uint32_t a, uint32_t b) {
  // Montgomery: returns a*b*R^{-1} mod P
  uint64_t t = (uint64_t)a * b;
  uint32_t m = (uint32_t)t * NPRIME;
  uint64_t u = (t + (uint64_t)m * P32) >> 32;
  uint32_t r = (uint32_t)u;
  return (r >= P32) ? r - P32 : r;
}
__device__ __forceinline__ uint32_t addm(uint32_t a, uint32_t b) {
  uint32_t s = a + b; return (s >= P32) ? s - P32 : s;
}
__device__ __forceinline__ uint32_t subm(uint32_t a, uint32_t b) {
  return (a >= b) ? a - b : a + P32 - b;
}
__device__ __forceinline__ uint32_t rev21(uint32_t i) { return __brev(i) >> (32 - LOG_N); }

// ---------------- twiddle table: tw[k] = mont(iroot^k), k < N/2 ----------------
__global__ void k_twiddles(uint32_t* __restrict__ twid) {
  uint32_t k = blockIdx.x * blockDim.x + threadIdx.x;
  if (k >= NN / 2) return;
  uint32_t acc = R1;                       // mont(1)
  uint32_t base = mmul(IROOT, R2);         // mont(iroot)
  uint32_t e = k;
  while (e) { if (e & 1) acc = mmul(acc, base); base = mmul(base, base); e >>= 1; }
  twid[k] = acc;
}

// ---------------- bit-reversed load of the 4 witness polys ----------------
struct Ptr4 { const int* p[4]; };
__global__ void k_bitrev_witness(Ptr4 src, uint32_t* __restrict__ polys) {
  uint32_t i = blockIdx.x * blockDim.x + threadIdx.x;
  if (i >= NN) return;
  uint32_t y = blockIdx.y;                 // poly slot 0..3
  polys[(size_t)y * NN + i] = (uint32_t)src.p[y][rev21(i)];
}

// ---------------- compress4 + lookup select + sort-buffer fill ----------------
__global__ void k_compress(const int* __restrict__ t1, const int* __restrict__ t2,
                           const int* __restrict__ t3, const int* __restrict__ t4,
                           const int* __restrict__ wl, const int* __restrict__ wr,
                           const int* __restrict__ wo, const int* __restrict__ w4,
                           const int* __restrict__ ql, const int* __restrict__ zeta,
                           uint32_t* __restrict__ t_arr, uint32_t* __restrict__ f_arr,
                           uint32_t* __restrict__ sbuf) {
  uint32_t i = blockIdx.x * blockDim.x + threadIdx.x;
  if (i >= NN) return;
  uint32_t z = (uint32_t)zeta[0]; if (z >= P32) z -= P32;
  uint32_t z1m = mmul(z, R2);              // mont(z)
  uint32_t z2m = mmul(z1m, z1m);           // mont(z^2)
  uint32_t z3m = mmul(z2m, z1m);           // mont(z^3)

  uint32_t tc = addm(addm((uint32_t)t1[i], mmul((uint32_t)t2[i], z1m)),
                     addm(mmul((uint32_t)t3[i], z2m), mmul((uint32_t)t4[i], z3m)));
  uint32_t wc = addm(addm((uint32_t)wl[i], mmul((uint32_t)wr[i], z1m)),
                     addm(mmul((uint32_t)wo[i], z2m), mmul((uint32_t)w4[i], z3m)));
  uint32_t fc = ql[i] ? wc : tc;

  uint32_t r = rev21(i);
  t_arr[r] = tc;                           // slot 4, bit-reversed for NTT
  f_arr[r] = fc;                           // slot 5
  sbuf[i] = tc;                            // concat for sort
  sbuf[NN + i] = fc;
}

// ---------------- bitonic sort (2^22 u32), LDS-fused low strides ----------------
__global__ void k_bitonic_local_full(uint32_t* __restrict__ d) {   // k = 2..2048
  __shared__ uint32_t sh[2048];
  uint32_t base = blockIdx.x * 2048u;
  uint32_t t = threadIdx.x;                // 1024
  sh[t] = d[base + t]; sh[t + 1024] = d[base + t + 1024];
  for (uint32_t k = 2; k <= 2048; k <<= 1) {
    for (uint32_t j = k >> 1; j > 0; j >>= 1) {
      __syncthreads();
      uint32_t i = ((t & ~(j - 1)) << 1) | (t & (j - 1));
      uint32_t q = i | j;
      bool dir = (((base + i) & k) == 0);
      uint32_t a = sh[i], b = sh[q];
      if ((a > b) == dir) { sh[i] = b; sh[q] = a; }
    }
  }
  __syncthreads();
  d[base + t] = sh[t]; d[base + t + 1024] = sh[t + 1024];
}

__global__ void k_bitonic_shared(uint32_t* __restrict__ d, uint32_t k) { // j = 1024..1
  __shared__ uint32_t sh[2048];
  uint32_t base = blockIdx.x * 2048u;
  uint32_t t = threadIdx.x;
  sh[t] = d[base + t]; sh[t + 1024] = d[base + t + 1024];
  for (uint32_t j = 1024; j > 0; j >>= 1) {
    __syncthreads();
    uint32_t i = ((t & ~(j - 1)) << 1) | (t & (j - 1));
    uint32_t q = i | j;
    bool dir = (((base + i) & k) == 0);
    uint32_t a = sh[i], b = sh[q];
    if ((a > b) == dir) { sh[i] = b; sh[q] = a; }
  }
  __syncthreads();
  d[base + t] = sh[t]; d[base + t + 1024] = sh[t + 1024];
}

__global__ void k_bitonic_global(uint32_t* __restrict__ d, uint32_t k, uint32_t j) {
  uint32_t t = blockIdx.x * blockDim.x + threadIdx.x;      // NTOT/2 threads
  uint32_t i = ((t & ~(j - 1)) << 1) | (t & (j - 1));
  uint32_t q = i | j;
  bool dir = ((i & k) == 0);
  uint32_t a = d[i], b = d[q];
  if ((a > b) == dir) { d[i] = b; d[q] = a; }
}

// ---------------- split sorted buffer into h1/h2 (bit-reversed) ----------------
__global__ void k_split(const uint32_t* __restrict__ sbuf, uint32_t* __restrict__ polys) {
  uint32_t i = blockIdx.x * blockDim.x + threadIdx.x;
  if (i >= NN) return;
  uint32_t r = rev21(i);
  polys[(size_t)6 * NN + r] = sbuf[2 * i];       // h1
  polys[(size_t)7 * NN + r] = sbuf[2 * i + 1];   // h2
}

// ---------------- NTT: LDS-fused stages 1..11 (lengths 2..2048) ----------------
// Tile load uses the CDNA5 async memory->LDS path (ASYNCcnt-tracked), so the
// 8 KB tile never bounces through VGPRs.
__global__ void k_ntt_lds(uint32_t* __restrict__ polys, const uint32_t* __restrict__ twid) {
  __shared__ uint32_t sh[2048];
  uint32_t* x = polys + (size_t)blockIdx.y * NN + (size_t)blockIdx.x * 2048u;
  uint32_t t = threadIdx.x;                // 1024 threads, all lanes active

  // async copy: each lane moves 8 bytes; VDST VGPR carries the LDS byte offset
  {
    uint32_t ldsoff = t * 8u;
    uint64_t gaddr = (uint64_t)(uintptr_t)(x + (size_t)t * 2u);
    asm volatile("global_load_async_to_lds_b64 %0, %1, off"
                 :: "v"(ldsoff), "v"(gaddr) : "memory");
    asm volatile("s_wait_asynccnt 0x0" ::: "memory");
  }
  __syncthreads();

  #pragma unroll 1
  for (int s = 1; s <= 11; ++s) {
    uint32_t half = 1u << (s - 1);
    uint32_t grp = t >> (s - 1);
    uint32_t j = t & (half - 1);
    uint32_t idx = (grp << s) + j;
    uint32_t tw = twid[j << (LOG_N - s)];
    uint32_t u = sh[idx];
    uint32_t v = mmul(sh[idx + half], tw);
    sh[idx] = addm(u, v);
    sh[idx + half] = subm(u, v);
    __syncthreads();
  }
  x[t] = sh[t]; x[t + 1024] = sh[t + 1024];
}

// ---------------- NTT: global stages 12..21 ----------------
__global__ void k_ntt_stage(uint32_t* __restrict__ polys, const uint32_t* __restrict__ twid, int s) {
  uint32_t t = blockIdx.x * blockDim.x + threadIdx.x;   // NN/2 threads
  uint32_t* x = polys + (size_t)blockIdx.y * NN;
  uint32_t half = 1u << (s - 1);
  uint32_t grp = t >> (s - 1);
  uint32_t j = t & (half - 1);
  uint32_t idx = (grp << s) + j;
  uint32_t tw = twid[j << (LOG_N - s)];
  uint32_t u = x[idx];
  uint32_t v = mmul(x[idx + half], tw);
  x[idx] = addm(u, v);
  x[idx + half] = subm(u, v);
}

// ---------------- finalize: scale by N^{-1}, emit float polys ----------------
__global__ void k_finalize(uint32_t* __restrict__ polys, float* __restrict__ out_polys) {
  uint32_t i = blockIdx.x * blockDim.x + threadIdx.x;
  if (i >= NN) return;
  size_t p = blockIdx.y;
  uint32_t v = mmul(polys[p * NN + i], NINV_MONT);
  polys[p * NN + i] = v;
  out_polys[p * NN + i] = (float)v;
}

// ---------------- commit: 8xN * Nx3 fp32 GEMM via V_WMMA_F32_16X16X4_F32 -------
typedef __attribute__((ext_vector_type(2))) float v2f;
typedef __attribute__((ext_vector_type(8))) float v8f;
#define CBLK 256

// A (16x4 f32) layout per ISA 7.12.2: lanes 0-15 -> M=lane, VGPR0/1 = K0/K1;
// lanes 16-31 -> M=lane-16, VGPR0/1 = K2/K3. B (4x16) assumed mirrored.
// Branchless: pointers are clamped to valid rows and dead lanes are zeroed by
// a multiplicative mask, so there is no EXEC divergence around the loads.
__global__ void k_commit_partial(const uint32_t* __restrict__ polys,
                                 const float* __restrict__ bases,
                                 float* __restrict__ partial) {
  int lane = threadIdx.x;                  // 32 = one wave, EXEC all-ones
  int M = lane & 15;
  int kof = (lane >> 4) << 1;              // 0 or 2
  int c = lane & 15;
  float maskA = (M < 8) ? (float)(1.0 / 2013265921.0) : 0.0f;  // folds 1/P
  float maskB = (c < 3) ? 1.0f : 0.0f;
  const uint32_t* pa = polys + (size_t)(M & 7) * NN + kof;     // always valid
  const float* pb = bases + (size_t)((c < 3) ? c : 0);         // always valid

  v8f acc = {};
  const int chunk = NN / CBLK;             // 8192 K-values per block
  int n0 = blockIdx.x * chunk;
  for (int n = n0; n < n0 + chunk; n += 16) {      // 4 WMMAs / iteration
    __builtin_prefetch(pa + n + 64, 0, 0);         // global_prefetch_b8
    __builtin_prefetch(pb + (size_t)(n + 64) * 3, 0, 0);
    #pragma unroll
    for (int u = 0; u < 4; ++u) {
      int nk = n + 4 * u;
      v2f a, b;
      a.x = (float)pa[nk] * maskA;
      a.y = (float)pa[nk + 1] * maskA;
      b.x = pb[(size_t)(nk + kof) * 3] * maskB;
      b.y = pb[(size_t)(nk + kof + 1) * 3] * maskB;
      acc = __builtin_amdgcn_wmma_f32_16x16x4_f32(false, a, false, b,
                                                  (short)0, acc, false, false);
    }
  }
  float* out = partial + (size_t)blockIdx.x * 256 + (size_t)lane * 8;
  #pragma unroll
  for (int r = 0; r < 8; ++r) out[r] = acc[r];
}

__global__ void k_commit_reduce(const float* __restrict__ partial, float* __restrict__ out) {
  int t = threadIdx.x;                     // 256
  float s = 0.0f;
  for (int b = 0; b < CBLK; ++b) s += partial[(size_t)b * 256 + t];
  int lane = t >> 3, r = t & 7;
  int M = (lane < 16) ? r : r + 8;         // C/D 16x16 f32 layout
  int c = lane & 15;
  if (M < 8 && c < 3) out[M * 3 + c] = s;
}

// ---------------- host driver ----------------
extern "C" void kernel_launch(void* const* d_in, const int* in_sizes, int n_in,
                              void* d_out, int out_size, void* d_ws, size_t ws_size,
                              hipStream_t stream) {
  (void)in_sizes; (void)n_in; (void)out_size; (void)ws_size;
  const int* w_l = (const int*)d_in[0];
  const int* w_r = (const int*)d_in[1];
  const int* w_o = (const int*)d_in[2];
  const int* w_4 = (const int*)d_in[3];
  const int* t1  = (const int*)d_in[4];
  const int* t2  = (const int*)d_in[5];
  const int* t3  = (const int*)d_in[6];
  const int* t4  = (const int*)d_in[7];
  const int* ql  = (const int*)d_in[8];
  const float* bases = (const float*)d_in[9];
  const int* zeta = (const int*)d_in[10];

  float* out = (float*)d_out;              // [0..23] commits, [24..) 8*N polys

  // workspace layout: 8 polys (u32) | twiddles N/2 | sort buf 2N | commit partials
  uint32_t* polys = (uint32_t*)d_ws;
  uint32_t* twid  = polys + (size_t)8 * NN;
  uint32_t* sbuf  = twid + NN / 2;
  float* partial  = (float*)(sbuf + (size_t)2 * NN);

  // 1. twiddle table (Montgomery form)
  k_twiddles<<<(NN / 2) / 256, 256, 0, stream>>>(twid);

  // 2. bit-reversed witness loads (slots 0..3)
  Ptr4 wsrc; wsrc.p[0] = w_l; wsrc.p[1] = w_r; wsrc.p[2] = w_o; wsrc.p[3] = w_4;
  k_bitrev_witness<<<dim3(NN / 256, 4), 256, 0, stream>>>(wsrc, polys);

  // 3. compress (slots 4/5) + fill sort buffer
  k_compress<<<NN / 256, 256, 0, stream>>>(t1, t2, t3, t4, w_l, w_r, w_o, w_4,
                                           ql, zeta,
                                           polys + (size_t)4 * NN,
                                           polys + (size_t)5 * NN, sbuf);

  // 4. bitonic sort of 2N values
  k_bitonic_local_full<<<NTOT / 2048, 1024, 0, stream>>>(sbuf);
  for (uint32_t k = 4096; k <= NTOT; k <<= 1) {
    for (uint32_t j = k >> 1; j >= 2048; j >>= 1)
      k_bitonic_global<<<(NTOT / 2) / 1024, 1024, 0, stream>>>(sbuf, k, j);
    k_bitonic_shared<<<NTOT / 2048, 1024, 0, stream>>>(sbuf, k);
  }

  // 5. split into h1/h2 (slots 6/7)
  k_split<<<NN / 256, 256, 0, stream>>>(sbuf, polys);

  // 6. 8 batched inverse NTTs: LDS stages 1..11, then global stages 12..21
  k_ntt_lds<<<dim3(NN / 2048, 8), 1024, 0, stream>>>(polys, twid);
  for (int s = 12; s <= LOG_N; ++s)
    k_ntt_stage<<<dim3((NN / 2) / 1024, 8), 1024, 0, stream>>>(polys, twid, s);

  // 7. scale by N^{-1}, emit float polys
  k_finalize<<<dim3(NN / 256, 8), 256, 0, stream>>>(polys, out + 24);

  // 8. commit GEMM (8xN x Nx3) on the WMMA f32 path, deterministic 2-stage reduce
  k_commit_partial<<<CBLK, 32, 0, stream>>>(polys, bases, partial);
  k_commit_reduce<<<1, 256, 0, stream>>>(partial, out);
}